// CascadeEncoderFirst_32134945308969
// MI455X (gfx1250) — compile-verified
//
#include <hip/hip_runtime.h>
#include <math.h>

// ---------------------------------------------------------------------------
// CascadeEncoderFirst for MI455X (gfx1250), fp32 end-to-end.
// All GEMM-shaped compute uses V_WMMA_F32_16X16X4_F32.
// FFT low-pass implemented as low = R@X@R + I@X@I with Dirichlet matrices.
// Group-conv + 1x1 collapsed into one dense 3x3 conv with effective weights.
// B/X LDS tiles stored K-pair-interleaved so each WMMA B-fragment is one b64 load.
// ---------------------------------------------------------------------------

typedef __attribute__((ext_vector_type(2))) float v2f;
typedef __attribute__((ext_vector_type(4))) float v4f;
typedef __attribute__((ext_vector_type(8))) float v8f;

// Only check / call the builtin in the DEVICE pass (host pass has no amdgcn builtins).
#if defined(__gfx1250__) && !__has_builtin(__builtin_amdgcn_wmma_f32_16x16x4_f32)
#error "wmma f32 16x16x4 builtin missing on gfx1250 device pass"
#endif

#define NEG_SLOPE 0.01f
#define IN_EPS 1e-5f
static constexpr int HW = 256;
static constexpr int PLANE = HW * HW;          // 65536
static constexpr size_t TSZ = (size_t)4 * 64 * PLANE; // 16,777,216 elems per tensor
static constexpr int HL = 38;                  // int(256*0.3 // 2); freqs -38..37

// LDS tile strides (floats).
static constexpr int LDA_S = 34;   // A/W tile: 64 rows x 32 K (row-major, 8B-aligned rows)
static constexpr int LDBI  = 130;  // B/X tile: 16 K-pair rows x (64 n x 2) interleaved

__device__ __forceinline__ float lrelu(float v) { return v > 0.f ? v : NEG_SLOPE * v; }

__device__ __forceinline__ v8f wmma_f32_16x16x4(v2f a, v2f b, v8f c) {
#if defined(__gfx1250__)
  return __builtin_amdgcn_wmma_f32_16x16x4_f32(false, a, false, b, (short)0, c,
                                               false, false);
#else
  (void)a; (void)b;
  return c;  // host-pass stub, never executed
#endif
}

// One 16x16x4 fp32 WMMA step.
// A (16x4 MxK): lanes 0-15 -> K=(kk,kk+1); lanes 16-31 -> K=(kk+2,kk+3); M=lane&15.
// B (4x16 KxN): same K split per half-wave; fragment = interleaved pair -> one b64.
__device__ __forceinline__ v8f wmma_step(const float* As, const float* Bs,
                                         int msub, int nsub, int kk, int lane, v8f acc) {
  int half = lane >> 4;
  int kq = kk + (half << 1);
  int m = msub + (lane & 15);
  int n = nsub + (lane & 15);
  v2f a = *(const v2f*)&As[m * LDA_S + kq];            // kq even -> 8B aligned
  v2f b = *(const v2f*)&Bs[(kq >> 1) * LDBI + n * 2];  // (B[kq][n], B[kq+1][n])
  return wmma_f32_16x16x4(a, b, acc);
}

// ---------------------------------------------------------------------------
// Generic batched dense GEMM: C[img] (+)= A[img] @ B[img]
// M,N multiples of 64; K multiple of 32; lda/ldb multiples of 4. 256 thr = 8 waves.
// Block tile 64x64; wave (wv) covers M-sub (wv&3)*16 and N-sub (wv>>2)*32 (two 16x16).
// Global tiles fetched as b128 into distinct registers (batched under one wait).
// ---------------------------------------------------------------------------
__global__ __launch_bounds__(256) void gemm_wmma(
    const float* __restrict__ A, const float* __restrict__ B, float* __restrict__ C,
    int N, int K, int lda, int ldb, int ldc,
    long sA, long sB, long sC, int accumulate) {
  __shared__ float As[64 * LDA_S];
  __shared__ float Bs[16 * LDBI];
  int img = blockIdx.y;
  const float* Ab = A + (size_t)img * sA;
  const float* Bb = B + (size_t)img * sB;
  float* Cb = C + (size_t)img * sC;
  int tilesN = N >> 6;
  int m0 = (blockIdx.x / tilesN) << 6;
  int n0 = (blockIdx.x % tilesN) << 6;
  int tid = threadIdx.x, lane = tid & 31, wv = tid >> 5;
  int msub = (wv & 3) << 4, nsub = (wv >> 2) << 5;
  v8f c0 = {0.f, 0.f, 0.f, 0.f, 0.f, 0.f, 0.f, 0.f};
  v8f c1 = c0;
  for (int k0 = 0; k0 < K; k0 += 32) {
    v4f ra[2], rb[2];
#pragma unroll
    for (int i = 0; i < 2; ++i) {               // A: 64x32 = 512 v4f
      int e = tid + (i << 8);
      int r = e >> 3, c4 = (e & 7) << 2;
      ra[i] = *(const v4f*)&Ab[(size_t)(m0 + r) * lda + k0 + c4];
    }
#pragma unroll
    for (int i = 0; i < 2; ++i) {               // B: 32x64 = 512 v4f
      int e = tid + (i << 8);
      int r = e >> 4, c4 = (e & 15) << 2;
      rb[i] = *(const v4f*)&Bb[(size_t)(k0 + r) * ldb + n0 + c4];
    }
    if (k0 + 32 < K) {                          // prefetch next K tile
      __builtin_prefetch(&Ab[(size_t)(m0 + (tid >> 4)) * lda + k0 + 32], 0, 1);
      __builtin_prefetch(&Bb[(size_t)(k0 + 32 + (tid >> 5)) * ldb + n0], 0, 1);
    }
#pragma unroll
    for (int i = 0; i < 2; ++i) {               // A row-major store (2 x b64)
      int e = tid + (i << 8);
      int r = e >> 3, c4 = (e & 7) << 2;
      v2f lo = {ra[i].x, ra[i].y}, hi = {ra[i].z, ra[i].w};
      *(v2f*)&As[r * LDA_S + c4] = lo;
      *(v2f*)&As[r * LDA_S + c4 + 2] = hi;
    }
#pragma unroll
    for (int i = 0; i < 2; ++i) {               // B K-pair-interleaved store
      int e = tid + (i << 8);
      int r = e >> 4, c4 = (e & 15) << 2;
      int kp = r >> 1, lo = r & 1;
      float* dst = &Bs[kp * LDBI + c4 * 2 + lo];
      dst[0] = rb[i].x; dst[2] = rb[i].y; dst[4] = rb[i].z; dst[6] = rb[i].w;
    }
    __syncthreads();
#pragma unroll
    for (int kk = 0; kk < 32; kk += 4) {
      c0 = wmma_step(As, Bs, msub, nsub, kk, lane, c0);
      c1 = wmma_step(As, Bs, msub, nsub + 16, kk, lane, c1);
    }
    __syncthreads();
  }
  int col = n0 + nsub + (lane & 15);
  int rbase = m0 + msub + ((lane >> 4) << 3);
#pragma unroll
  for (int r = 0; r < 8; ++r) {
    int row = rbase + r;
    size_t i0 = (size_t)row * ldc + col;
    float v0 = c0[r], v1 = c1[r];
    if (accumulate) { v0 += Cb[i0]; v1 += Cb[i0 + 16]; }
    Cb[i0] = v0;
    Cb[i0 + 16] = v1;
  }
}

// ---------------------------------------------------------------------------
// Implicit-GEMM conv (3x3 pad=1, or 1x1): out[b,co,h,w] = sum_k W[co,k]*im2col + bias (+res)
// Input channels may be split across two NCHW tensors (virtual concat).
// Block computes 64 out-channels x 64 consecutive pixels of one row.
// K = (ch0+ch1)*taps, zero-padded to multiple of 32 inside the K loop.
// ---------------------------------------------------------------------------
__global__ __launch_bounds__(256) void conv_gemm_wmma(
    const float* __restrict__ in0, int ch0,
    const float* __restrict__ in1, int ch1,
    const float* __restrict__ W, int taps,
    const float* __restrict__ bias,
    const float* __restrict__ res,
    float* __restrict__ out) {
  int Cin = ch0 + ch1;
  int K = Cin * taps;
  __shared__ float Ws[64 * LDA_S];
  __shared__ float Xs[16 * LDBI];
  int b = blockIdx.y;
  int h = blockIdx.x >> 2;
  int w0 = (blockIdx.x & 3) << 6;
  int tid = threadIdx.x, lane = tid & 31, wv = tid >> 5;
  int msub = (wv & 3) << 4, nsub = (wv >> 2) << 5;
  v8f c0 = {0.f, 0.f, 0.f, 0.f, 0.f, 0.f, 0.f, 0.f};
  v8f c1 = c0;
  for (int k0 = 0; k0 < K; k0 += 32) {
    float rw[8], rx[8];
#pragma unroll
    for (int i = 0; i < 8; ++i) {               // weights 64x32
      int e = tid + (i << 8);
      int r = e >> 5, c = e & 31;
      int k = k0 + c;
      rw[i] = (k < K) ? W[(size_t)r * K + k] : 0.f;
    }
#pragma unroll
    for (int i = 0; i < 8; ++i) {               // im2col 32x64
      int e = tid + (i << 8);
      int r = e >> 6, n = e & 63;
      int k = k0 + r;
      float v = 0.f;
      if (k < K) {
        int c, t;
        if (taps == 1) { c = k; t = 0; } else { c = k / 9; t = k - 9 * c; }
        int dy = (taps == 9) ? (t / 3) - 1 : 0;
        int dx = (taps == 9) ? (t % 3) - 1 : 0;
        int hh = h + dy, ww = w0 + n + dx;
        if (hh >= 0 && hh < HW && ww >= 0 && ww < HW) {
          const float* base = (c < ch0)
              ? (in0 + ((size_t)b * ch0 + c) * PLANE)
              : (in1 + ((size_t)b * ch1 + (c - ch0)) * PLANE);
          v = base[hh * HW + ww];
        }
      }
      rx[i] = v;
    }
#pragma unroll
    for (int i = 0; i < 8; ++i) {
      int e = tid + (i << 8);
      Ws[(e >> 5) * LDA_S + (e & 31)] = rw[i];
    }
#pragma unroll
    for (int i = 0; i < 8; ++i) {               // K-pair-interleaved X store
      int e = tid + (i << 8);
      int r = e >> 6, n = e & 63;
      Xs[(r >> 1) * LDBI + n * 2 + (r & 1)] = rx[i];
    }
    __syncthreads();
#pragma unroll
    for (int kk = 0; kk < 32; kk += 4) {
      c0 = wmma_step(Ws, Xs, msub, nsub, kk, lane, c0);
      c1 = wmma_step(Ws, Xs, msub, nsub + 16, kk, lane, c1);
    }
    __syncthreads();
  }
  int col = w0 + nsub + (lane & 15);
  int rbase = msub + ((lane >> 4) << 3);
#pragma unroll
  for (int r = 0; r < 8; ++r) {
    int co = rbase + r;
    size_t idx = ((size_t)b * 64 + co) * PLANE + (size_t)h * HW + col;
    float v0 = c0[r], v1 = c1[r];
    if (bias) { float bb = bias[co]; v0 += bb; v1 += bb; }
    if (res) { v0 += res[idx]; v1 += res[idx + 16]; }
    out[idx] = v0;
    out[idx + 16] = v1;
  }
}

// --- Dirichlet (ideal low-pass) matrices: R[i,j] + i*I[i,j] = (1/256) sum_{k=-38}^{37} e^{i 2pi k (i-j)/256}
__global__ __launch_bounds__(256) void build_dirichlet(float* __restrict__ R, float* __restrict__ I) {
  int idx = blockIdx.x * 256 + threadIdx.x;   // 65536 entries
  int i = idx >> 8, j = idx & 255;
  double d = (double)(i - j);
  double w = 6.283185307179586476925286766559 / 256.0 * d;
  double cs = 0.0, sn = 0.0;
  for (int k = -HL; k < HL; ++k) {
    double a = w * (double)k;
    cs += cos(a);
    sn += sin(a);
  }
  R[idx] = (float)(cs / 256.0);
  I[idx] = (float)(sn / 256.0);
}

// --- Collapse grouped conv (w_con: 512x1x3x3, groups=128) + 1x1 (w_u1: 64x512)
//     into Weff[co, g, t] = sum_{j<4} w_u1[co, 4g+j] * w_con[4g+j, t]
__global__ __launch_bounds__(256) void build_weff(const float* __restrict__ w_u1,
                                                  const float* __restrict__ w_con,
                                                  float* __restrict__ weff) {
  int idx = blockIdx.x * 256 + threadIdx.x;   // 64*128*9 = 73728
  if (idx >= 64 * 128 * 9) return;
  int co = idx / 1152;
  int r = idx - co * 1152;
  int g = r / 9, t = r - g * 9;
  float s = 0.f;
  for (int j = 0; j < 4; ++j)
    s += w_u1[(size_t)co * 512 + 4 * g + j] * w_con[(size_t)(4 * g + j) * 9 + t];
  weff[idx] = s;
}

// --- per-(b,c) spatial mean (AdaptiveAvgPool2d(1))
__global__ __launch_bounds__(256) void channel_mean(const float* __restrict__ x,
                                                    float* __restrict__ m) {
  __shared__ double sd[256];
  int p = blockIdx.x;                         // 256 planes
  const float* base = x + (size_t)p * PLANE;
  double s = 0.0;
  for (int i = threadIdx.x; i < PLANE; i += 256) s += (double)base[i];
  sd[threadIdx.x] = s;
  __syncthreads();
  for (int o = 128; o > 0; o >>= 1) {
    if (threadIdx.x < o) sd[threadIdx.x] += sd[threadIdx.x + o];
    __syncthreads();
  }
  if (threadIdx.x == 0) m[p] = (float)(sd[0] / (double)PLANE);
}

// --- squeeze-excite MLP: a = sigmoid(W2 @ lrelu(W1 @ mean)); one block per batch
__global__ __launch_bounds__(64) void attention_mlp(const float* __restrict__ m,
                                                    const float* __restrict__ w1,
                                                    const float* __restrict__ w2,
                                                    float* __restrict__ scale) {
  __shared__ float a1[16];
  int b = blockIdx.x, t = threadIdx.x;
  if (t < 16) {
    float s = 0.f;
    for (int c = 0; c < 64; ++c) s += w1[t * 64 + c] * m[b * 64 + c];
    a1[t] = lrelu(s);
  }
  __syncthreads();
  float s = 0.f;
  for (int j = 0; j < 16; ++j) s += w2[t * 16 + j] * a1[j];
  scale[b * 64 + t] = 1.f / (1.f + expf(-s));
}

// --- depthwise 3x3 refine of (xp * scale): xw = dw3x3(xp*a, w_ref) + b_ref
__global__ __launch_bounds__(256) void dw_refine(const float* __restrict__ xp,
                                                 const float* __restrict__ scale,
                                                 const float* __restrict__ w,
                                                 const float* __restrict__ bias,
                                                 float* __restrict__ out) {
  size_t idx = (size_t)blockIdx.x * 256 + threadIdx.x;   // 16,777,216
  int p = (int)(idx >> 16);
  int c = p & 63;
  int hw = (int)(idx & (PLANE - 1));
  int h = hw >> 8, wq = hw & 255;
  float sc = scale[p];
  const float* base = xp + (size_t)p * PLANE;
  const float* wk = w + c * 9;
  float acc = bias[c];
  for (int t = 0; t < 9; ++t) {
    int hh = h + t / 3 - 1, ww = wq + t % 3 - 1;
    if (hh >= 0 && hh < HW && ww >= 0 && ww < HW)
      acc += wk[t] * base[hh * HW + ww] * sc;
  }
  out[idx] = acc;
}

// --- InstanceNorm statistics (biased var), double accumulation
__global__ __launch_bounds__(256) void inorm_stats(const float* __restrict__ x,
                                                   float* __restrict__ stats) {
  __shared__ double sd[256];
  __shared__ double sq[256];
  int p = blockIdx.x;
  const float* base = x + (size_t)p * PLANE;
  double s = 0.0, q = 0.0;
  for (int i = threadIdx.x; i < PLANE; i += 256) {
    double v = (double)base[i];
    s += v;
    q += v * v;
  }
  sd[threadIdx.x] = s;
  sq[threadIdx.x] = q;
  __syncthreads();
  for (int o = 128; o > 0; o >>= 1) {
    if (threadIdx.x < o) {
      sd[threadIdx.x] += sd[threadIdx.x + o];
      sq[threadIdx.x] += sq[threadIdx.x + o];
    }
    __syncthreads();
  }
  if (threadIdx.x == 0) {
    double mean = sd[0] / (double)PLANE;
    double var = sq[0] / (double)PLANE - mean * mean;
    stats[p] = (float)mean;
    stats[256 + p] = (float)(1.0 / sqrt(var + (double)IN_EPS));
  }
}

__global__ __launch_bounds__(256) void inorm_apply_lrelu(float* __restrict__ x,
                                                         const float* __restrict__ stats) {
  size_t idx = (size_t)blockIdx.x * 256 + threadIdx.x;
  int p = (int)(idx >> 16);
  float v = (x[idx] - stats[p]) * stats[256 + p];
  x[idx] = lrelu(v);
}

// --- 2x2 max pool
__global__ __launch_bounds__(256) void maxpool2(const float* __restrict__ x,
                                                float* __restrict__ y) {
  size_t idx = (size_t)blockIdx.x * 256 + threadIdx.x;   // 4*64*128*128
  int p = (int)(idx >> 14);
  int hw = (int)(idx & 16383);
  int h = hw >> 7, w = hw & 127;
  const float* base = x + (size_t)p * PLANE + (size_t)(2 * h) * HW + 2 * w;
  float a = base[0], b = base[1], c = base[HW], d = base[HW + 1];
  float m = a > b ? a : b;
  float n = c > d ? c : d;
  y[idx] = m > n ? m : n;
}

// ---------------------------------------------------------------------------
extern "C" void kernel_launch(void* const* d_in, const int* in_sizes, int n_in,
                              void* d_out, int out_size, void* d_ws, size_t ws_size,
                              hipStream_t stream) {
  (void)in_sizes; (void)n_in; (void)out_size; (void)ws_size;
  const float* x      = (const float*)d_in[0];
  const float* w_proj = (const float*)d_in[1];
  const float* b_proj = (const float*)d_in[2];
  const float* w_a1   = (const float*)d_in[3];
  const float* w_a2   = (const float*)d_in[4];
  const float* w_ref  = (const float*)d_in[5];
  const float* b_ref  = (const float*)d_in[6];
  const float* w_fuse = (const float*)d_in[7];
  const float* b_fuse = (const float*)d_in[8];
  const float* w_con  = (const float*)d_in[9];
  const float* w_u1   = (const float*)d_in[10];
  const float* b_u1   = (const float*)d_in[11];
  const float* w_u2   = (const float*)d_in[12];
  const float* b_u2   = (const float*)d_in[13];
  const float* w_sc   = (const float*)d_in[14];
  const float* b_sc   = (const float*)d_in[15];

  float* ws = (float*)d_ws;
  float* xp   = ws;             // proj output; later reused as u1 buffer
  float* xw   = ws + TSZ;       // refined high branch (xc lower half)
  float* t1   = ws + 2 * TSZ;   // R@X; later reused as lowf (xc upper half)
  float* t2   = ws + 3 * TSZ;   // I@X
  float* lowb = ws + 4 * TSZ;   // low; later reused as u2 buffer
  float* Rm   = ws + 5 * TSZ;
  float* Im   = Rm + PLANE;
  float* Weff = Im + PLANE;
  float* stats = Weff + 64 * 128 * 9;
  float* amean = stats + 512;
  float* ascl  = amean + 256;

  float* feat = (float*)d_out;
  float* pooled = feat + TSZ;

  dim3 blk(256);
  dim3 convGrid(1024, 4);       // 256 rows * 4 col-tiles per image, 4 images
  dim3 gemmGrid(16, 256);       // 4x4 64-tiles, 256 (b,c) images
  const int ELT_BLKS = (int)(TSZ / 256);       // 65536
  const int POOL_BLKS = (int)(TSZ / 4 / 256);  // 16384

  // Constant precompute
  build_dirichlet<<<256, blk, 0, stream>>>(Rm, Im);
  build_weff<<<288, blk, 0, stream>>>(w_u1, w_con, Weff);

  // xp = conv3x3(x, w_proj) + b_proj   (K=27, WMMA implicit GEMM)
  conv_gemm_wmma<<<convGrid, blk, 0, stream>>>(x, 3, nullptr, 0, w_proj, 9,
                                               b_proj, nullptr, xp);

  // Channel attention on detached xp
  channel_mean<<<256, blk, 0, stream>>>(xp, amean);
  attention_mlp<<<4, 64, 0, stream>>>(amean, w_a1, w_a2, ascl);

  // High branch: depthwise refine of (xp * a), then InstanceNorm + LeakyReLU
  dw_refine<<<ELT_BLKS, blk, 0, stream>>>(xp, ascl, w_ref, b_ref, xw);
  inorm_stats<<<256, blk, 0, stream>>>(xw, stats);
  inorm_apply_lrelu<<<ELT_BLKS, blk, 0, stream>>>(xw, stats);

  // Low branch: ideal low-pass == R@X@R + I@X@I (batched WMMA GEMMs over 256 planes)
  gemm_wmma<<<gemmGrid, blk, 0, stream>>>(Rm, xp, t1, 256, 256, 256, 256, 256,
                                          0, PLANE, PLANE, 0);
  gemm_wmma<<<gemmGrid, blk, 0, stream>>>(Im, xp, t2, 256, 256, 256, 256, 256,
                                          0, PLANE, PLANE, 0);
  gemm_wmma<<<gemmGrid, blk, 0, stream>>>(t1, Rm, lowb, 256, 256, 256, 256, 256,
                                          PLANE, 0, PLANE, 0);
  gemm_wmma<<<gemmGrid, blk, 0, stream>>>(t2, Im, lowb, 256, 256, 256, 256, 256,
                                          PLANE, 0, PLANE, 1);

  // lowf = lrelu(inorm(conv1x1(low, w_fuse)+b))   (reuse t1)
  conv_gemm_wmma<<<convGrid, blk, 0, stream>>>(lowb, 64, nullptr, 0, w_fuse, 1,
                                               b_fuse, nullptr, t1);
  inorm_stats<<<256, blk, 0, stream>>>(t1, stats);
  inorm_apply_lrelu<<<ELT_BLKS, blk, 0, stream>>>(t1, stats);

  // u1 = lrelu(inorm(conv3x3(concat(xw,lowf), Weff)+b_u1))  (K=1152; reuse xp)
  conv_gemm_wmma<<<convGrid, blk, 0, stream>>>(xw, 64, t1, 64, Weff, 9,
                                               b_u1, nullptr, xp);
  inorm_stats<<<256, blk, 0, stream>>>(xp, stats);
  inorm_apply_lrelu<<<ELT_BLKS, blk, 0, stream>>>(xp, stats);

  // u2 = lrelu(inorm(conv3x3(u1, w_u2)+b_u2))   (K=576; reuse lowb)
  conv_gemm_wmma<<<convGrid, blk, 0, stream>>>(xp, 64, nullptr, 0, w_u2, 9,
                                               b_u2, nullptr, lowb);
  inorm_stats<<<256, blk, 0, stream>>>(lowb, stats);
  inorm_apply_lrelu<<<ELT_BLKS, blk, 0, stream>>>(lowb, stats);

  // feat = u2 + conv1x1(concat(xw,lowf), w_sc)+b_sc  (K=128, fused residual)
  conv_gemm_wmma<<<convGrid, blk, 0, stream>>>(xw, 64, t1, 64, w_sc, 1,
                                               b_sc, lowb, feat);

  // pooled = maxpool2(feat)
  maxpool2<<<POOL_BLKS, blk, 0, stream>>>(feat, pooled);
}